// BVHEncoder_72954314489880
// MI455X (gfx1250) — compile-verified
//
#include <hip/hip_runtime.h>
#include <cstdint>

typedef float v2f __attribute__((ext_vector_type(2)));
typedef float v8f __attribute__((ext_vector_type(8)));

// Cross-half lane swap (lane ^= 16) via LDS swizzle unit (no LDS memory touched).
__device__ __forceinline__ float swapx16(float v) {
  return __builtin_bit_cast(float,
      __builtin_amdgcn_ds_swizzle(__builtin_bit_cast(int, v), 0x401f));
}

// One wave = 32 rays (2 WMMA column tiles of 16) vs all 64 leaves (4 row tiles).
// Center/extent slab form via V_WMMA_F32_16X16X4_F32:
//   tc[l,r] = (lmin+lmax)[l,d] * (0.5*inv[r,d]) + (-orig*inv)[r,d]   A=(sum,1)
//   te[l,r] = (lmax-lmin)[l,d] * |0.5*inv[r,d]|  >= 0                A=(dif,0)
//   lo = tc - te, hi = tc + te      (no abs/canonicalize on D needed at all)
// Leaf-tile loop kept rolled to bound live WMMA results (<128 VGPRs).
__global__ void __launch_bounds__(256)
bvh_leaf_wmma_kernel(const float* __restrict__ x,
                     const float* __restrict__ leaf_min,
                     const float* __restrict__ leaf_max,
                     float* __restrict__ out, int n) {
  // [0,192) leaf_min, [192,384) leaf_max, [384,768) zeros (upper-lane A source)
  __shared__ float s_big[768];

  const int tid = threadIdx.x;

  // ---- Stage leaf AABBs to LDS with CDNA5 async-to-LDS loads (ASYNCcnt) ----
  if (tid < 192) {
    uint32_t lds0 = (uint32_t)(uintptr_t)&s_big[tid];
    uint32_t lds1 = (uint32_t)(uintptr_t)&s_big[192 + tid];
    int boff = tid * 4;
    asm volatile("global_load_async_to_lds_b32 %0, %1, %2"
                 :: "v"(lds0), "v"(boff), "s"(leaf_min) : "memory");
    asm volatile("global_load_async_to_lds_b32 %0, %1, %2"
                 :: "v"(lds1), "v"(boff), "s"(leaf_max) : "memory");
  }
  s_big[384 + tid] = 0.0f;                    // zero block for lanes 16-31
  if (tid < 128) s_big[640 + tid] = 0.0f;
  asm volatile("s_wait_asynccnt 0" ::: "memory");
  __syncthreads();

  const int lane = tid & 31;
  const bool lo16 = (lane & 16) == 0;
  const int gid = blockIdx.x * 256 + tid;     // one ray per lane

  // ---- Per-ray setup: origin, normalized dir, 0.5*inv, -orig*inv ----
  const float2* xp = (const float2*)(x + (size_t)gid * 6);
  float2 q0 = xp[0], q1 = xp[1], q2 = xp[2];
  float o[3]   = {q0.x, q0.y, q1.x};
  float dxs[3] = {q1.y - o[0], q2.x - o[1], q2.y - o[2]};
  float rn = 1.0f / sqrtf(dxs[0]*dxs[0] + dxs[1]*dxs[1] + dxs[2]*dxs[2]);
  float v[3], hiv[3], cc[3], shiv[3], scc[3];
  #pragma unroll
  for (int d = 0; d < 3; ++d) {
    v[d]   = dxs[d] * rn;
    float iv = 1.0f / v[d];
    hiv[d] = 0.5f * iv;
    cc[d]  = -o[d] * iv;
    shiv[d] = swapx16(hiv[d]);  // other half-wave's rays, for column tile 1
    scc[d]  = swapx16(cc[d]);
  }

  // Lanes 0-15 read real leaf data (K=0,1 rows); lanes 16-31 read the zero
  // block, which supplies the mandatory K=2,3 zeros with no cndmask at all.
  const float* abase = &s_big[lo16 ? 0 : 384];
  const int lrow = (lane & 15) * 3;
  const float aone = lo16 ? 1.0f : 0.0f;
  const float INF  = __builtin_inff();
  const float TINY = 1.4012984643e-45f;       // FLT_TRUE_MIN: tf>0 <=> tf>=TINY
  const v8f czero = {};

  #pragma unroll
  for (int rt = 0; rt < 2; ++rt) {            // two 16-ray column tiles
    v2f Bc[3], Be[3];
    #pragma unroll
    for (int d = 0; d < 3; ++d) {
      float bx = rt ? shiv[d] : hiv[d];
      float by = rt ? scc[d]  : cc[d];
      Bc[d].x = bx;         Bc[d].y = by;     // K=0: 0.5/v, K=1: -o/v
      Be[d].x = fabsf(bx);  Be[d].y = by;     // K=0: |0.5/v| (K=1 unused: A.y=0)
    }

    v8f bestkey, bestt2;
    #pragma unroll
    for (int j = 0; j < 8; ++j) { bestkey[j] = INF; bestt2[j] = 0.0f; }

    #pragma unroll 1                          // keep rolled: bounds VGPR pressure
    for (int lt = 0; lt < 4; ++lt) {
      const float* pm = abase + lt * 48 + lrow;
      v8f tnear = czero, tfar = czero;
      #pragma unroll
      for (int d = 0; d < 3; ++d) {
        float mn = pm[d];                     // leaf_min (or 0 in upper lanes)
        float mx = pm[192 + d];               // leaf_max (or 0 in upper lanes)
        v2f Ac, Ae;
        Ac.x = mn + mx;  Ac.y = aone;
        Ae.x = mx - mn;  Ae.y = 0.0f;
        v8f tc = __builtin_amdgcn_wmma_f32_16x16x4_f32(
            false, Ac, false, Bc[d], (short)0, czero, false, false);
        v8f te = __builtin_amdgcn_wmma_f32_16x16x4_f32(
            false, Ae, false, Be[d], (short)0, czero, false, false);
        #pragma unroll
        for (int j = 0; j < 8; ++j) {
          float lo = tc[j] - te[j];           // te >= 0 by construction
          float hi = tc[j] + te[j];
          if (d == 0) { tnear[j] = lo; tfar[j] = hi; }
          else {
            tnear[j] = fmaxf(tnear[j], lo);
            tfar[j]  = fminf(tfar[j],  hi);
          }
        }
      }
      // hit test + nearest-hit tracking (elementwise over 8 leaves/lane)
      #pragma unroll
      for (int j = 0; j < 8; ++j) {
        float tn = tnear[j], tf = tfar[j];
        float gate = fmaxf(fmaxf(tn, 0.0f), TINY);   // v_max3
        bool hit = (tf >= gate);              // == (tf>=max(tn,0) && tf>0)
        float key = hit ? tn : INF;
        bool u = key < bestkey[j];
        bestkey[j] = u ? key : bestkey[j];
        bestt2[j]  = u ? tf  : bestt2[j];
      }
    }

    // reduce over the 8 leaves held in-lane, then merge the two leaf halves
    float bk = bestkey[0], bt = bestt2[0];
    #pragma unroll
    for (int j = 1; j < 8; ++j) {
      bool u = bestkey[j] < bk;
      bk = u ? bestkey[j] : bk;
      bt = u ? bestt2[j]  : bt;
    }
    float obk = swapx16(bk), obt = swapx16(bt);
    bool u2 = obk < bk;
    bk = u2 ? obk : bk;
    bt = u2 ? obt : bt;

    bool  m  = bk < INF;
    float t1 = m ? bk : 0.0f;
    float t2 = m ? bt : 0.0f;

    // ray of this tile's column (lane%16) is exactly this lane's own ray
    // when (lane>>4)==rt, so its o[]/v[] are already resident.
    if ((lane >> 4) == rt) {
      size_t nn = (size_t)n;
      float2* op = (float2*)(out + (size_t)gid * 6);
      float2 p0, p1, p2;
      p0.x = o[0] + v[0]*t1; p0.y = o[1] + v[1]*t1;
      p1.x = o[2] + v[2]*t1; p1.y = o[0] + v[0]*t2;
      p2.x = o[1] + v[1]*t2; p2.y = o[2] + v[2]*t2;
      op[0] = p0; op[1] = p1; op[2] = p2;
      out[6*nn + gid] = m ? 1.0f : 0.0f;
      out[7*nn + gid] = t1;
      out[8*nn + gid] = t2;
    }
  }
}

extern "C" void kernel_launch(void* const* d_in, const int* in_sizes, int n_in,
                              void* d_out, int out_size, void* d_ws, size_t ws_size,
                              hipStream_t stream) {
  const float* x    = (const float*)d_in[0];
  const float* lmin = (const float*)d_in[1];
  const float* lmax = (const float*)d_in[2];
  float* out = (float*)d_out;
  int n = in_sizes[0] / 6;        // 262144 rays (multiple of 256)
  int blocks = n / 256;           // 8 waves/block, 32 rays/wave
  bvh_leaf_wmma_kernel<<<blocks, 256, 0, stream>>>(x, lmin, lmax, out, n);
}